// Attention_47837345743361
// MI455X (gfx1250) — compile-verified
//
#include <hip/hip_runtime.h>
#include <math.h>

// ---------------------------------------------------------------------------
// Attention over 65536 candidates, D=2048, fused one-pass (flash) softmax.
// MI455X (gfx1250): wave32, WMMA f32 16x16x4 for the score GEMV.
// ---------------------------------------------------------------------------

typedef __attribute__((ext_vector_type(2)))  float    v2f;
typedef __attribute__((ext_vector_type(8)))  float    v8f;

constexpr int DIM          = 2048;    // embedding dim
constexpr int TILE_ROWS    = 16;      // rows per wave-tile (WMMA M)
constexpr int WAVE         = 32;      // gfx1250 is wave32
constexpr int BLOCK        = 256;     // 8 waves / block
constexpr int WPB          = BLOCK / WAVE;
constexpr int PART_STRIDE  = DIM + 4; // [m, s, pad, pad, vec(2048)] -> 16B-aligned vec
constexpr int KSPLIT       = 8;       // K-split for the align GEMV

// ---------------------------------------------------------------------------
// Kernel 1a: partial align GEMV. Both words fused (one 16MB pass over W_a),
// K split across blockIdx.y so 64 blocks stream W_a instead of 8. Loads of
// W_a[d*DIM + j] are perfectly coalesced; word[d] is a scalar broadcast.
// Deterministic partial sums (no FP atomics).
// ---------------------------------------------------------------------------
__global__ void align_partial_kernel(const float* __restrict__ wl,
                                     const float* __restrict__ wr,
                                     const float* __restrict__ Wa,
                                     float* __restrict__ ap) {
  const int j  = blockIdx.x * blockDim.x + threadIdx.x;   // column
  const int kb = blockIdx.y;                              // K-split index
  const int d0 = kb * (DIM / KSPLIT);
  float accl = 0.f, accr = 0.f;
  const float* __restrict__ col = Wa + (size_t)d0 * DIM + j;
  for (int d = 0; d < DIM / KSPLIT; ++d) {
    const float w = col[(size_t)d * DIM];
    accl = fmaf(wl[d0 + d], w, accl);
    accr = fmaf(wr[d0 + d], w, accr);
  }
  ap[((size_t)kb * 2 + 0) * DIM + j] = accl;
  ap[((size_t)kb * 2 + 1) * DIM + j] = accr;
}

// Kernel 1b: reduce K-split partials, add bias, tanh.
__global__ void align_finish_kernel(const float* __restrict__ ap,
                                    const float* __restrict__ ba,
                                    float* __restrict__ align_buf) {
  const int j = blockIdx.x * blockDim.x + threadIdx.x;
  float accl = 0.f, accr = 0.f;
#pragma unroll
  for (int kb = 0; kb < KSPLIT; ++kb) {
    accl += ap[((size_t)kb * 2 + 0) * DIM + j];
    accr += ap[((size_t)kb * 2 + 1) * DIM + j];
  }
  align_buf[j]       = tanhf(accl + ba[j]);
  align_buf[DIM + j] = tanhf(accr + ba[j]);
}

// ---------------------------------------------------------------------------
// Kernel 2: one-pass flash attention reduction.
//   side 0: candidates_r weighted by softmax(candidates_r @ align_l) -> out_l
//   side 1: candidates_l weighted by softmax(candidates_l @ align_r) -> out_r
// Each wave owns 16-row tiles (strided by P across the grid) and keeps
// (running max m, running sum s, running weighted vec[2048]) state.
// Scores come from V_WMMA_F32_16X16X4_F32:
//   A: 16x4 f32 tile of candidate rows (lanes 0-15: K=k,k+1; 16-31: K=k+2,k+3)
//   B: every lane of a half-wave holds the SAME align chunk, so all 16 B
//      columns are identical -> every D column redundantly holds the score
//      vector. This removes the per-fragment EXEC masking/divergent LDS load
//      entirely (no saveexec, no cndmask in the hot loop).
//   D: lane 0 holds scores M=0..7 (acc[0..7]), lane 16 holds M=8..15.
// k-loop is batched 8 fragments at a time so 8 global b64 + 8 broadcast
// ds b64 issue before the chained WMMAs -> real memory-level parallelism.
// The weighted accumulation re-reads the 128KB tile just streamed for the
// score pass (hits WGP$/L2), keeping HBM traffic at ~1x per candidate
// matrix (~1GB total => ~43us at 23.3 TB/s).
// ---------------------------------------------------------------------------
__global__ void flash_kernel(const float* __restrict__ cand_l,
                             const float* __restrict__ cand_r,
                             const float* __restrict__ align_buf, // [2*DIM]
                             float* __restrict__ partials,        // [2*P*PART_STRIDE]
                             int P, int N) {
  const int side = blockIdx.y;
  const float* __restrict__ cand = (side == 0) ? cand_r : cand_l;

  __shared__ float lds_align[DIM];
  {
    const float* __restrict__ src = align_buf + side * DIM;
    for (int i = threadIdx.x; i < DIM; i += blockDim.x) lds_align[i] = src[i];
  }
  __syncthreads();

  const int lane  = threadIdx.x & (WAVE - 1);
  const int wave  = threadIdx.x / WAVE;
  const int gwave = blockIdx.x * WPB + wave;          // 0..P-1
  const int r16   = lane & 15;
  const int hi    = lane >> 4;                        // 0 or 1
  const int koff  = hi << 1;                          // K sub-offset 0 or 2

  float  m    = -INFINITY;
  float  ssum = 0.f;
  float4 vec[16];                                     // 64 f32 of vec state/lane
#pragma unroll
  for (int i = 0; i < 16; ++i) vec[i] = float4{0.f, 0.f, 0.f, 0.f};

  const int ntiles = N / TILE_ROWS;
  for (int tile = gwave; tile < ntiles; tile += P) {
    const int row0 = tile * TILE_ROWS;
    const float* __restrict__ rowp = cand + (size_t)(row0 + r16) * DIM;

    // prefetch the next tile's rows into cache (global_prefetch_b8)
    if (tile + P < ntiles) {
      const float* pf =
          cand + (size_t)((tile + P) * TILE_ROWS + r16) * DIM + hi * 1024;
      __builtin_prefetch(pf, 0, 0);
    }

    // ---------------- scores via WMMA ----------------
    v8f acc = {};
    const float* __restrict__ ap = rowp + koff;
    for (int k = 0; k < DIM; k += 32) {
      float2 af[8], bf[8];
#pragma unroll
      for (int u = 0; u < 8; ++u) {
        af[u] = *(const float2*)(ap + k + 4 * u);               // global b64
        bf[u] = *(const float2*)&lds_align[k + 4 * u + koff];   // ds b64 bcast
      }
#pragma unroll
      for (int u = 0; u < 8; ++u) {
        v2f a; a[0] = af[u].x; a[1] = af[u].y;
        v2f b; b[0] = bf[u].x; b[1] = bf[u].y;
        acc = __builtin_amdgcn_wmma_f32_16x16x4_f32(
            false, a, false, b, (short)0, acc, false, false);
      }
    }

    // extract the 16 scores (lane 0: M=0..7, lane 16: M=8..15), broadcast
    float sc[16];
#pragma unroll
    for (int i = 0; i < 8; ++i) {
      sc[i]     = __shfl(acc[i], 0, WAVE);
      sc[8 + i] = __shfl(acc[i], 16, WAVE);
    }

    // ---------------- online softmax update ----------------
    float mt = m;
#pragma unroll
    for (int i = 0; i < 16; ++i) mt = fmaxf(mt, sc[i]);
    const float scale = __expf(m - mt);                // 0 on first tile (m=-inf)
    float w[16];
    float wsum = 0.f;
#pragma unroll
    for (int i = 0; i < 16; ++i) { w[i] = __expf(sc[i] - mt); wsum += w[i]; }
    ssum = ssum * scale + wsum;
    m    = mt;
#pragma unroll
    for (int i = 0; i < 16; ++i) {
      vec[i].x *= scale; vec[i].y *= scale; vec[i].z *= scale; vec[i].w *= scale;
    }

    // ---------------- weighted accumulation (tile re-read from cache) ------
    for (int r = 0; r < TILE_ROWS; ++r) {
      const float* __restrict__ rp = cand + (size_t)(row0 + r) * DIM;
      const float wr_ = w[r];
#pragma unroll
      for (int c = 0; c < 16; ++c) {
        const float4 cv = *(const float4*)(rp + c * 128 + lane * 4); // b128
        vec[c].x = fmaf(wr_, cv.x, vec[c].x);
        vec[c].y = fmaf(wr_, cv.y, vec[c].y);
        vec[c].z = fmaf(wr_, cv.z, vec[c].z);
        vec[c].w = fmaf(wr_, cv.w, vec[c].w);
      }
    }
  }

  // ---------------- write this wave's partial ----------------
  float* __restrict__ pb = partials + ((size_t)side * P + gwave) * PART_STRIDE;
  if (lane == 0) { pb[0] = m; pb[1] = ssum; }
  float* __restrict__ pv = pb + 4;                     // 16B aligned
#pragma unroll
  for (int c = 0; c < 16; ++c)
    *(float4*)(pv + c * 128 + lane * 4) = vec[c];
}

// ---------------------------------------------------------------------------
// Kernel 3: merge partials with log-sum-exp weights and normalize.
// Thread d owns output column d; (m,s) loads are uniform scalar reads, vec
// loads are coalesced. out = [out_l(2048) | out_r(2048)].
// ---------------------------------------------------------------------------
__global__ void combine_kernel(const float* __restrict__ partials,
                               int P, float* __restrict__ out) {
  const int side = blockIdx.y;
  const int d    = blockIdx.x * blockDim.x + threadIdx.x;
  const float* __restrict__ pb = partials + (size_t)side * P * PART_STRIDE;

  float M = -INFINITY;
  for (int p = 0; p < P; ++p) M = fmaxf(M, pb[(size_t)p * PART_STRIDE]);

  float S = 0.f, acc = 0.f;
  for (int p = 0; p < P; ++p) {
    const float* __restrict__ row = pb + (size_t)p * PART_STRIDE;
    const float wp = __expf(row[0] - M);
    S   = fmaf(row[1], wp, S);
    acc = fmaf(wp, row[4 + d], acc);
  }
  out[(size_t)side * DIM + d] = acc / S;
}

// ---------------------------------------------------------------------------
extern "C" void kernel_launch(void* const* d_in, const int* in_sizes, int n_in,
                              void* d_out, int out_size, void* d_ws, size_t ws_size,
                              hipStream_t stream) {
  const float* wl = (const float*)d_in[0];   // embed_word_l        [1,D]
  const float* wr = (const float*)d_in[1];   // embed_word_r        [1,D]
  const float* cl = (const float*)d_in[2];   // embed_candidates_l  [N,D]
  const float* cr = (const float*)d_in[3];   // embed_candidates_r  [N,D]
  const float* Wa = (const float*)d_in[4];   // W_a                 [D,D]
  const float* ba = (const float*)d_in[5];   // b_a                 [1,D]
  float* out = (float*)d_out;
  float* ws  = (float*)d_ws;

  const int N = in_sizes[2] / DIM;           // 65536

  // choose partial count P (waves per side) to fit workspace, deterministic
  int P = 2048;
  while (P > 8 &&
         (size_t)(2 * DIM + 2 * (size_t)P * PART_STRIDE + 2 * KSPLIT * DIM) *
                 sizeof(float) > ws_size)
    P >>= 1;

  float* align_buf = ws;                                // [2*DIM]
  float* partials  = ws + 2 * DIM;                      // [2*P*PART_STRIDE]
  float* align_ws  = partials + 2 * (size_t)P * PART_STRIDE; // [KSPLIT*2*DIM]

  align_partial_kernel<<<dim3(DIM / BLOCK, KSPLIT), dim3(BLOCK), 0, stream>>>(
      wl, wr, Wa, align_ws);

  align_finish_kernel<<<dim3(DIM / BLOCK), dim3(BLOCK), 0, stream>>>(
      align_ws, ba, align_buf);

  flash_kernel<<<dim3(P / WPB, 2), dim3(BLOCK), 0, stream>>>(
      cl, cr, align_buf, partials, P, N);

  combine_kernel<<<dim3(DIM / BLOCK, 2), dim3(BLOCK), 0, stream>>>(
      partials, P, out);
}